// FeatureVectorGraft_88510686036004
// MI455X (gfx1250) — compile-verified
//
#include <hip/hip_runtime.h>
#include <hip/hip_bf16.h>

typedef __attribute__((ext_vector_type(2))) float v2f;
typedef __attribute__((ext_vector_type(4))) float v4f;
typedef __attribute__((ext_vector_type(8))) float v8f;

#define BATCH 4
#define SEQ 4096
#define DMODEL 4096
#define DFEAT 256
#define LN_EPS 1e-5f
#define STRENGTH 1.0f

// Workspace layout (floats):
//   [0 .. 4095]   featsP : 16 x 256 zero-padded normalized features (A matrix)
//   [4096..4099]  scale  : per-batch applies flag (0.0 / 1.0)
//   [4100..4103]  last   : per-batch last-token index (as int)

// ---------------------------------------------------------------------------
// Stage 1: fused LayerNorm + trigger scan + mask-sum. One block per batch row.
// ---------------------------------------------------------------------------
__global__ void __launch_bounds__(256)
graft_prep_kernel(const float* __restrict__ ff,      // [B, DFEAT]
                  const int*   __restrict__ tok,     // [B, SEQ]
                  const int*   __restrict__ am,      // [B, SEQ]
                  const float* __restrict__ gamma,   // [DFEAT]
                  const float* __restrict__ beta,    // [DFEAT]
                  float* __restrict__ featsP,        // [16, DFEAT]
                  float* __restrict__ scale,         // [B]
                  int*   __restrict__ last)          // [B]
{
    const int b = blockIdx.x;      // 0..3
    const int t = threadIdx.x;     // 0..255

    __shared__ float red[256];
    __shared__ int   ired[256];
    __shared__ float s_mu, s_rstd;

    // --- trigger / mask reduction (packed: hits<<16 | msum; both sums safe) ---
    int hit = 0, msum = 0;
    for (int s = t; s < SEQ; s += 256) {
        int id = tok[b * SEQ + s];
        hit  += (id == 5) | (id == 7);
        msum += am[b * SEQ + s];
    }
    ired[t] = (hit << 16) | msum;
    __syncthreads();
    for (int off = 128; off > 0; off >>= 1) {
        if (t < off) ired[t] += ired[t + off];
        __syncthreads();
    }

    // --- LayerNorm: mean ---
    float x = ff[b * DFEAT + t];
    red[t] = x;
    __syncthreads();
    for (int off = 128; off > 0; off >>= 1) {
        if (t < off) red[t] += red[t + off];
        __syncthreads();
    }
    if (t == 0) s_mu = red[0] * (1.0f / DFEAT);
    __syncthreads();
    float mu = s_mu;

    // --- variance ---
    float d = x - mu;
    red[t] = d * d;
    __syncthreads();
    for (int off = 128; off > 0; off >>= 1) {
        if (t < off) red[t] += red[t + off];
        __syncthreads();
    }
    if (t == 0) s_rstd = rsqrtf(red[0] * (1.0f / DFEAT) + LN_EPS);
    __syncthreads();

    // normalized feature row (A-matrix row b)
    featsP[b * DFEAT + t] = d * s_rstd * gamma[t] + beta[t];
    // zero-pad rows 4..15 of the A matrix (3 rows per block)
    for (int j = 0; j < 3; ++j)
        featsP[(4 + b * 3 + j) * DFEAT + t] = 0.0f;

    if (t == 0) {
        int packed = ired[0];
        scale[b] = ((packed >> 16) != 0) ? STRENGTH : 0.0f;
        int ms = packed & 0xFFFF;
        last[b] = (ms < 1 ? 1 : ms) - 1;
    }
}

// ---------------------------------------------------------------------------
// Stage 2: streaming copy x -> out, non-temporal B128 traffic (512 MB total,
// ~22 us at 23.3 TB/s HBM; this is the roofline for the whole op).
// ---------------------------------------------------------------------------
__global__ void __launch_bounds__(256)
graft_copy_kernel(const float* __restrict__ x, float* __restrict__ out, long long nvec)
{
    const v4f* __restrict__ src = (const v4f*)x;
    v4f* __restrict__ dst = (v4f*)out;
    long long i = (long long)blockIdx.x * blockDim.x + threadIdx.x;
    long long stride = (long long)gridDim.x * blockDim.x;
    for (; i < nvec; i += stride) {
        v4f v = __builtin_nontemporal_load(src + i);
        __builtin_nontemporal_store(v, dst + i);
    }
}

// ---------------------------------------------------------------------------
// Stage 3: delta GEMM via V_WMMA_F32_16X16X4_F32 + masked scatter-add.
// One wave per 16-wide tile of DMODEL (256 waves total).
//   A (16x4 per step): A[m][k]  m = lane&15, k = kstep + (lane>>4)*2 + v
//   B (4x16 per step): B[k][n] = W[n][k], n = lane&15, same k indexing
//   C/D 16x16 f32: lanes 0..15 / VGPR v hold D[m=v][n=lane]; v<4 are batches.
// ---------------------------------------------------------------------------
__global__ void __launch_bounds__(256)
graft_wmma_kernel(const float* __restrict__ W,       // [DMODEL, DFEAT]
                  const float* __restrict__ bias,    // [DMODEL]
                  const float* __restrict__ featsP,  // [16, DFEAT]
                  const float* __restrict__ scale,   // [B]
                  const int*   __restrict__ last,    // [B]
                  float* __restrict__ out)           // [B, SEQ, DMODEL]
{
    const int wave  = (blockIdx.x * blockDim.x + threadIdx.x) >> 5;  // 0..255
    const int lane  = threadIdx.x & 31;
    const int nbase = wave * 16;
    const int mn    = lane & 15;
    const int koff  = (lane >> 4) * 2;

    const float* __restrict__ arow = featsP + (size_t)mn * DFEAT;
    const float* __restrict__ brow = W + (size_t)(nbase + mn) * DFEAT;

    v8f c = {};
#pragma unroll 4
    for (int kk = 0; kk < DFEAT; kk += 4) {
        v2f a = *(const v2f*)(arow + kk + koff);
        v2f b = *(const v2f*)(brow + kk + koff);
        c = __builtin_amdgcn_wmma_f32_16x16x4_f32(
                /*neg_a=*/false, a, /*neg_b=*/false, b,
                /*c_mod=*/(short)0, c, /*reuse_a=*/false, /*reuse_b=*/false);
    }

    if (lane < 16) {
        const int n = nbase + mn;
        const float bn = bias[n];
#pragma unroll
        for (int v = 0; v < BATCH; ++v) {
            float dlt = (c[v] + bn) * scale[v];
            size_t idx = ((size_t)v * SEQ + (size_t)last[v]) * DMODEL + n;
            out[idx] += dlt;
        }
    }
}

extern "C" void kernel_launch(void* const* d_in, const int* in_sizes, int n_in,
                              void* d_out, int out_size, void* d_ws, size_t ws_size,
                              hipStream_t stream) {
    const float* x     = (const float*)d_in[0];
    const float* ff    = (const float*)d_in[1];
    const int*   tok   = (const int*)d_in[2];
    const int*   am    = (const int*)d_in[3];
    const float* gamma = (const float*)d_in[4];
    const float* beta  = (const float*)d_in[5];
    const float* W     = (const float*)d_in[6];
    const float* bias  = (const float*)d_in[7];
    float* out = (float*)d_out;

    float* featsP = (float*)d_ws;            // 16*256 floats
    float* scale  = featsP + 16 * DFEAT;     // 4 floats
    int*   last   = (int*)(scale + BATCH);   // 4 ints

    // Stage 1: LN + trigger/mask scan
    graft_prep_kernel<<<BATCH, 256, 0, stream>>>(ff, tok, am, gamma, beta,
                                                 featsP, scale, last);

    // Stage 2: bulk copy (bandwidth-bound roofline: 512 MB @ 23.3 TB/s)
    long long nvec = (long long)BATCH * SEQ * DMODEL / 4;   // 16,777,216 float4
    graft_copy_kernel<<<4096, 256, 0, stream>>>(x, out, nvec);

    // Stage 3: WMMA delta + scatter-add (256 waves = 32 blocks * 8 waves)
    graft_wmma_kernel<<<32, 256, 0, stream>>>(W, bias, featsP, scale, last, out);
}